// CombinedSOTALoss_16329465659754
// MI455X (gfx1250) — compile-verified
//
#include <hip/hip_runtime.h>
#include <hip/hip_bf16.h>

// Problem dims (fixed by the reference harness).
#define BATCH   16
#define HH      1024
#define WW      1024
#define IMGPIX  (HH * WW)                       // 1<<20
#define NPIX    ((long long)BATCH * IMGPIX)     // 16777216

// Erosion tiling.
#define TW      128
#define TH      32
#define PITCH   (TW + 2)     // 130
#define TROWS   (TH + 2)     // 34
#define SENT    0xFFFFFFFFu

#if defined(__has_builtin)
#if __has_builtin(__builtin_amdgcn_tensor_load_to_lds) && __has_builtin(__builtin_amdgcn_s_wait_tensorcnt)
#define USE_TDM 1
#endif
#endif
#ifndef USE_TDM
#define USE_TDM 0
#endif

typedef __attribute__((ext_vector_type(4))) unsigned int u32x4_t;
typedef __attribute__((ext_vector_type(8))) int          i32x8_t;
typedef __attribute__((ext_vector_type(4))) int          i32x4_t;

// ---------------------------------------------------------------- init
__global__ void k_init(float* facc, unsigned int* uacc) {
    if (threadIdx.x == 0 && blockIdx.x == 0) {
        for (int i = 0; i < 16; ++i) facc[i] = 0.f;
        for (int i = 0; i < 20; ++i) uacc[i] = 0u;   // per-iteration sums
        uacc[20] = 0u; uacc[21] = 0u;                // done flags (pred, tgt)
        uacc[22] = SENT; uacc[23] = SENT;            // prev sums (== "inf")
    }
}

// ---------------------------------------------------------------- binarize
// binary_pred = sigmoid(pred) > 0.5  <=>  pred > 0 ;  binary_tgt = target > 0.5
__global__ __launch_bounds__(256) void k_binarize(
    const float* __restrict__ pred, const float* __restrict__ tgt,
    unsigned char* __restrict__ binP, unsigned char* __restrict__ binT,
    unsigned char* __restrict__ skP,  unsigned char* __restrict__ skT) {
    long long base = ((long long)blockIdx.x * 256 + threadIdx.x) * 16;
    __builtin_prefetch(pred + base + 4096, 0, 1);   // global_prefetch_b8
    __builtin_prefetch(tgt  + base + 4096, 0, 1);
    unsigned char bp[16] __attribute__((aligned(16)));
    unsigned char bt[16] __attribute__((aligned(16)));
    const float4* p4 = (const float4*)(pred + base);
    const float4* t4 = (const float4*)(tgt + base);
#pragma unroll
    for (int v = 0; v < 4; ++v) {
        float4 p = p4[v], t = t4[v];
        bp[4*v+0] = p.x > 0.f; bp[4*v+1] = p.y > 0.f;
        bp[4*v+2] = p.z > 0.f; bp[4*v+3] = p.w > 0.f;
        bt[4*v+0] = t.x > 0.5f; bt[4*v+1] = t.y > 0.5f;
        bt[4*v+2] = t.z > 0.5f; bt[4*v+3] = t.w > 0.5f;
    }
    uint4 vp = *(const uint4*)bp, vt = *(const uint4*)bt;
    *(uint4*)(binP + base) = vp;  *(uint4*)(skP + base) = vp;
    *(uint4*)(binT + base) = vt;  *(uint4*)(skT + base) = vt;
}

// ---------------------------------------------------------------- erosion pass
// new = (s - er <= 0 ? s : 0) + er,  er = (sum3x3 >= 9);  integer-exact vs ref.
__global__ __launch_bounds__(256) void k_erode(
    const unsigned char* __restrict__ src, unsigned char* __restrict__ dst,
    unsigned int* __restrict__ iterSum, const unsigned int* __restrict__ done) {
    __shared__ __align__(16) unsigned char tile[TROWS * PITCH];
    __shared__ unsigned int sred[256];
    const int t  = threadIdx.x;
    const int x0 = blockIdx.x * TW, y0 = blockIdx.y * TH, img = blockIdx.z;
    const long long ibase = (long long)img * IMGPIX;
    const int r  = t >> 3;            // 32 rows, 8 threads/row
    const int cb = (t & 7) * 16;      // 16 px per thread

    if (*done != 0u) {                // converged: skeleton passes through
        long long gi = ibase + (long long)(y0 + r) * WW + x0 + cb;
        *(uint4*)(dst + gi) = *(const uint4*)(src + gi);
        return;
    }

    bool coop = true;
#if USE_TDM
    if (x0 > 0 && y0 > 0) {           // TDM fast path; OOB zero-fills right/bottom halo
        coop = false;
        if (t < 32) {                 // wave 0 issues the tensor DMA (EXEC all-ones)
            unsigned lds = (unsigned)(unsigned long long)(uintptr_t)&tile[0];
            unsigned long long ga =
                (unsigned long long)(uintptr_t)(src + ibase + (long long)(y0 - 1) * WW + (x0 - 1));
            unsigned td0 = (unsigned)(WW - (x0 - 1));   // remaining width from tile start
            unsigned td1 = (unsigned)(HH - (y0 - 1));   // remaining height
            u32x4_t g0; i32x8_t g1; i32x4_t gz4; i32x8_t gz8;
            gz4[0] = 0; gz4[1] = 0; gz4[2] = 0; gz4[3] = 0;
            gz8[0] = 0; gz8[1] = 0; gz8[2] = 0; gz8[3] = 0;
            gz8[4] = 0; gz8[5] = 0; gz8[6] = 0; gz8[7] = 0;
            g0[0] = 1u;                                   // count=1, gather off
            g0[1] = lds;                                  // lds_addr
            g0[2] = (unsigned)ga;                         // global_addr[31:0]
            g0[3] = (unsigned)((ga >> 32) & 0x1FFFFFFull) | (2u << 30); // addr hi | type=2
            g1[0] = 0;                                    // data_size=1B, no pad/iter/mask
            g1[1] = (int)((td0 & 0xFFFFu) << 16);
            g1[2] = (int)((td0 >> 16) | ((td1 & 0xFFFFu) << 16));
            g1[3] = (int)((td1 >> 16) | ((unsigned)PITCH << 16));  // tile_dim0=130
            g1[4] = (int)TROWS;                                    // tile_dim1=34
            g1[5] = WW;                                            // dim0 stride
            g1[6] = 0; g1[7] = 0;
            __builtin_amdgcn_tensor_load_to_lds(g0, g1, gz4, gz4, gz8, 0);
            __builtin_amdgcn_s_wait_tensorcnt(0);
        }
        __syncthreads();
    }
#endif
    if (coop) {                       // edge blocks: bounds-checked cooperative fill
        for (int i = t; i < TROWS * PITCH; i += 256) {
            int rr = i / PITCH, cc = i % PITCH;
            int gy = y0 - 1 + rr, gx = x0 - 1 + cc;
            unsigned char v = 0;
            if ((unsigned)gy < (unsigned)HH && (unsigned)gx < (unsigned)WW)
                v = src[ibase + (long long)gy * WW + gx];
            tile[i] = v;
        }
        __syncthreads();
    }

    // Column sums -> 3x3 sums for 16 outputs per thread.
    int cs[18];
#pragma unroll
    for (int x = 0; x < 18; ++x) {
        int c = cb + x;
        cs[x] = (int)tile[r * PITCH + c] + (int)tile[(r + 1) * PITCH + c]
              + (int)tile[(r + 2) * PITCH + c];
    }
    unsigned char outb[16] __attribute__((aligned(16)));
    unsigned int lsum = 0;
#pragma unroll
    for (int i = 0; i < 16; ++i) {
        int sum9 = cs[i] + cs[i + 1] + cs[i + 2];
        int s    = (int)tile[(r + 1) * PITCH + cb + 1 + i];
        int er   = (sum9 >= 9) ? 1 : 0;
        int nv   = ((s - er) <= 0 ? s : 0) + er;
        outb[i]  = (unsigned char)nv;
        lsum    += (unsigned)nv;
    }
    long long go = ibase + (long long)(y0 + r) * WW + x0 + cb;
    *(uint4*)(dst + go) = *(const uint4*)outb;

    sred[t] = lsum;
    __syncthreads();
    for (int o = 128; o > 0; o >>= 1) { if (t < o) sred[t] += sred[t + o]; __syncthreads(); }
    if (t == 0) atomicAdd(iterSum, sred[0]);
}

// ---------------------------------------------------------------- convergence flag
__global__ void k_update(const unsigned int* curr, unsigned int* prev, unsigned int* done) {
    if (*done == 0u) {
        unsigned c = *curr;
        if (*prev != SENT && c == *prev) *done = 1u;  // |curr - prev| < 1 (integers)
        *prev = c;
    }
}

// ---------------------------------------------------------------- fused loss pass
__global__ __launch_bounds__(256) void k_final(
    const float* __restrict__ pred, const float* __restrict__ tgt,
    const unsigned char* __restrict__ binP, const unsigned char* __restrict__ binT,
    const unsigned char* __restrict__ skP,  const unsigned char* __restrict__ skT,
    float* __restrict__ parts, int nblocks) {
    __shared__ float sh[256];
    const int t = threadIdx.x;
    long long idx = (long long)blockIdx.x * 256 + t;
    int img = (int)(idx >> 20);
    int rc  = (int)(idx & (IMGPIX - 1));
    int r = rc >> 10, c = rc & (WW - 1);
    long long ib = (long long)img << 20;
    __builtin_prefetch(pred + idx + WW, 0, 1);

    float p  = 1.f / (1.f + __expf(-pred[idx]));
    float tv = tgt[idx];

    float pn = 0.f, tn = 0.f, gxP = 0.f, gyP = 0.f, gxT = 0.f, gyT = 0.f;
#pragma unroll
    for (int dy = -1; dy <= 1; ++dy)
#pragma unroll
        for (int dx = -1; dx <= 1; ++dx) {
            int yy = r + dy, xx = c + dx;
            float pv = 0.f, tv2 = 0.f; int bp = 0, bt = 0;
            if ((unsigned)yy < (unsigned)HH && (unsigned)xx < (unsigned)WW) {
                long long j = ib + (long long)yy * WW + xx;
                pv = 1.f / (1.f + __expf(-pred[j]));
                tv2 = tgt[j]; bp = binP[j]; bt = binT[j];
            }
            pn += pv; tn += tv2;
            float sxw = (float)(dx * ((dy == 0) ? 2 : 1));   // sobel_x
            float syw = (float)(dy * ((dx == 0) ? 2 : 1));   // sobel_y
            gxP += sxw * bp; gyP += syw * bp;
            gxT += sxw * bt; gyT += syw * bt;
        }
    pn *= (1.f / 9.f); tn *= (1.f / 9.f);

    float eP = __fsqrt_rn(gxP * gxP + gyP * gyP + 1e-8f);
    float eT = __fsqrt_rn(gxT * gxT + gyT * gyT + 1e-8f);
    float ps = fminf(fmaxf((float)skP[idx] + 0.5f * eP, 0.f), 1.f);
    float ts = fminf(fmaxf((float)skT[idx] + 0.5f * eT, 0.f), 1.f);

    float pc  = fminf(fmaxf(p, 1e-6f), 1.f - 1e-6f);
    float bce = -(tv * __logf(pc) + (1.f - tv) * __logf(1.f - pc));
    float pt  = (tv == 1.f) ? pc : (1.f - pc);
    float fo  = ((tv == 1.f) ? 0.25f : 0.75f) * (1.f - pt) * (1.f - pt) * bce;

    float wm = (tv > 0.5f) ? 1.f : 0.f;
    float dv = pn - tn;

    float vals[9] = { ps * ts, ps, ts, p * tv, p, tv, fo, dv * dv * wm, wm };
#pragma unroll 1
    for (int k = 0; k < 9; ++k) {
        sh[t] = vals[k];
        __syncthreads();
        for (int o = 128; o > 0; o >>= 1) { if (t < o) sh[t] += sh[t + o]; __syncthreads(); }
        if (t == 0) parts[(long long)k * nblocks + blockIdx.x] = sh[0];
        __syncthreads();
    }
}

// Deterministic second-stage reduction (no float atomics).
__global__ __launch_bounds__(256) void k_reduce(const float* __restrict__ parts,
                                                int nblocks, float* __restrict__ facc) {
    __shared__ float sh[256];
    const int t = threadIdx.x;
    for (int k = 0; k < 9; ++k) {
        float s = 0.f;
        for (int b = t; b < nblocks; b += 256) s += parts[(long long)k * nblocks + b];
        sh[t] = s;
        __syncthreads();
        for (int o = 128; o > 0; o >>= 1) { if (t < o) sh[t] += sh[t + o]; __syncthreads(); }
        if (t == 0) facc[k] = sh[0];
        __syncthreads();
    }
}

__global__ void k_finalize(const float* facc, float* out) {
    float inter_s = facc[0], pssum = facc[1], tssum = facc[2];
    float inter   = facc[3], psum  = facc[4], tsum  = facc[5];
    float fsum    = facc[6], csum  = facc[7], nwater = facc[8];
    float skeleton = 1.f - (2.f * inter_s + 1.f) / (pssum + tssum + 1.f);
    float dice     = 1.f - (2.f * inter   + 1.f) / (psum  + tsum  + 1.f);
    float focal    = fminf(fmaxf(fsum / (float)NPIX, 0.f), 10.f);
    float conn     = (nwater == 0.f) ? 0.f : csum / fmaxf(nwater, 1.f);
    float total = 0.3f * skeleton + 0.4f * dice + 0.2f * focal + 0.1f * conn;
    if (isnan(total) || isinf(total)) total = dice;
    out[0] = total;
}

// ---------------------------------------------------------------- host launcher
extern "C" void kernel_launch(void* const* d_in, const int* in_sizes, int n_in,
                              void* d_out, int out_size, void* d_ws, size_t ws_size,
                              hipStream_t stream) {
    const float* pred = (const float*)d_in[0];
    const float* tgt  = (const float*)d_in[1];
    unsigned char* ws8 = (unsigned char*)d_ws;

    unsigned char* BIN_P = ws8;                     // u8 [NPIX]
    unsigned char* BIN_T = ws8 + 1 * NPIX;
    unsigned char* SK_PA = ws8 + 2 * NPIX;          // pred skel ping (final lands here)
    unsigned char* SK_B  = ws8 + 3 * NPIX;          // shared pong
    unsigned char* SK_TA = ws8 + 4 * NPIX;          // tgt skel ping (final lands here)
    float*        facc   = (float*)(ws8 + 5 * NPIX);
    unsigned int* uacc   = (unsigned int*)(facc + 16);
    float*        parts  = (float*)(ws8 + 5 * NPIX + 4096);
    const int nblk_final = (int)(NPIX / 256);       // 65536

    k_init<<<1, 64, 0, stream>>>(facc, uacc);
    k_binarize<<<(int)(NPIX / (256 * 16)), 256, 0, stream>>>(
        pred, tgt, BIN_P, BIN_T, SK_PA, SK_TA);

    dim3 eg(WW / TW, HH / TH, BATCH);
    for (int i = 0; i < 10; ++i) {                  // pred skeleton
        const unsigned char* s = (i & 1) ? SK_B : SK_PA;
        unsigned char*       d = (i & 1) ? SK_PA : SK_B;
        k_erode<<<eg, 256, 0, stream>>>(s, d, &uacc[i], &uacc[20]);
        k_update<<<1, 1, 0, stream>>>(&uacc[i], &uacc[22], &uacc[20]);
    }
    for (int i = 0; i < 10; ++i) {                  // target skeleton (reuses pong)
        const unsigned char* s = (i & 1) ? SK_B : SK_TA;
        unsigned char*       d = (i & 1) ? SK_TA : SK_B;
        k_erode<<<eg, 256, 0, stream>>>(s, d, &uacc[10 + i], &uacc[21]);
        k_update<<<1, 1, 0, stream>>>(&uacc[10 + i], &uacc[23], &uacc[21]);
    }

    k_final<<<nblk_final, 256, 0, stream>>>(pred, tgt, BIN_P, BIN_T, SK_PA, SK_TA,
                                            parts, nblk_final);
    k_reduce<<<1, 256, 0, stream>>>(parts, nblk_final, facc);
    k_finalize<<<1, 1, 0, stream>>>(facc, (float*)d_out);
}